// TranVectorQuantizer_35459249996161
// MI455X (gfx1250) — compile-verified
//
#include <hip/hip_runtime.h>

typedef __attribute__((ext_vector_type(2))) float v2f;
typedef __attribute__((ext_vector_type(4))) float v4f;
typedef __attribute__((ext_vector_type(8))) float v8f;

#define N_ROWS   131072      // 16384 * 8
#define EMB_DIM  32
#define K_CODES  128

// ---------------------------------------------------------------------------
// Kernel 1: VQ core. One wave32 handles 16 rows. x(16x32) @ cb^T(32x128)
// via 64 x v_wmma_f32_16x16x4_f32 per wave, then argmin + gather + STE.
// ---------------------------------------------------------------------------
__global__ __launch_bounds__(256) void vq_wmma_kernel(
    const float* __restrict__ latent,   // [N_ROWS, 32]
    const float* __restrict__ cb,       // [128, 32]
    float* __restrict__ policy,         // [N_ROWS, 32]
    float* __restrict__ quant)          // [N_ROWS, 32]
{
    const int lane    = threadIdx.x & 31;
    const int waveId  = blockIdx.x * 8 + (threadIdx.x >> 5);
    const int rowBase = waveId * 16;
    const int half    = lane >> 4;     // 0 or 1
    const int col     = lane & 15;

    // ---- Preload A tile: lane l holds {x[row][4s+2h], x[row][4s+2h+1]} ----
    v2f a[8];
    {
        const int row = rowBase + col;
        #pragma unroll
        for (int s = 0; s < 8; ++s)
            a[s] = *(const v2f*)(latent + row * EMB_DIM + 4 * s + 2 * half);
    }

    float bestVal[8];
    int   bestIdx[8];
    #pragma unroll
    for (int v = 0; v < 8; ++v) { bestVal[v] = 3.4e38f; bestIdx[v] = 0x7fffffff; }

    // ---- 8 output tiles of 16 codes each ----
    #pragma unroll
    for (int t = 0; t < 8; ++t) {
        const int n = t * 16 + col;          // code index this lane covers
        v8f acc = {0.f, 0.f, 0.f, 0.f, 0.f, 0.f, 0.f, 0.f};
        float cn_part = 0.f;
        #pragma unroll
        for (int s = 0; s < 8; ++s) {
            v2f b = *(const v2f*)(cb + n * EMB_DIM + 4 * s + 2 * half);
            cn_part += b.x * b.x + b.y * b.y;
            // D = A * B + C, full fp32
            acc = __builtin_amdgcn_wmma_f32_16x16x4_f32(
                false, a[s], false, b, (short)0, acc, false, false);
        }
        // ||c_n||^2: this half summed half the k's; partner half has the rest
        float cnorm = cn_part + __shfl_xor(cn_part, 16, 32);

        #pragma unroll
        for (int v = 0; v < 8; ++v) {
            // score = ||c||^2 - 2*dot  (||x||^2 constant per row, drop it)
            float d = cnorm - 2.0f * acc[v];
            if (d < bestVal[v] || (d == bestVal[v] && n < bestIdx[v])) {
                bestVal[v] = d; bestIdx[v] = n;
            }
        }
    }

    // ---- min-reduce (value, first-index) within each 16-lane half ----
    #pragma unroll
    for (int m = 8; m >= 1; m >>= 1) {
        #pragma unroll
        for (int v = 0; v < 8; ++v) {
            float ov = __shfl_xor(bestVal[v], m, 32);
            int   oi = __shfl_xor(bestIdx[v], m, 32);
            if (ov < bestVal[v] || (ov == bestVal[v] && oi < bestIdx[v])) {
                bestVal[v] = ov; bestIdx[v] = oi;
            }
        }
    }

    // ---- gather + straight-through output; each half writes its 8 rows ----
    // C/D layout: VGPR v -> row (v + 8*half); all 16 lanes of a half agree on bestIdx.
    #pragma unroll
    for (int v = 0; v < 8; ++v) {
        const int row = rowBase + v + 8 * half;
        const int bi  = bestIdx[v];
        v2f q = *(const v2f*)(cb + bi * EMB_DIM + 2 * col);
        v2f L = *(const v2f*)(latent + row * EMB_DIM + 2 * col);
        v2f p;
        p.x = L.x + (q.x - L.x);   // match jnp expression exactly
        p.y = L.y + (q.y - L.y);
        *(v2f*)(quant  + row * EMB_DIM + 2 * col) = q;
        *(v2f*)(policy + row * EMB_DIM + 2 * col) = p;
    }
}

// ---------------------------------------------------------------------------
// Kernel 2: broadcast codebook -> [16384, 128, 32] (256 MB). Pure store-BW.
// Stage 16 KB codebook in LDS, then non-temporal 128-bit streaming stores.
// ---------------------------------------------------------------------------
__global__ __launch_bounds__(256) void cb_broadcast_kernel(
    const float* __restrict__ cb,   // [128, 32]
    float* __restrict__ out)        // [16384, 128, 32]
{
    __shared__ v4f scb[1024];       // 128*32 floats = 1024 float4
    const int tid = threadIdx.x;
    const v4f* cb4 = (const v4f*)cb;
    #pragma unroll
    for (int i = 0; i < 4; ++i)
        scb[tid + 256 * i] = cb4[tid + 256 * i];
    __syncthreads();

    const unsigned long long total4 = 16777216ull;  // 67,108,864 floats / 4
    unsigned long long idx   = (unsigned long long)blockIdx.x * 256u + tid;
    const unsigned long long stride = (unsigned long long)gridDim.x * 256u;
    v4f* o4 = (v4f*)out;
    for (; idx < total4; idx += stride) {
        v4f val = scb[idx & 1023u];        // (flat float idx % 4096) / 4
        __builtin_nontemporal_store(val, o4 + idx);
    }
}

// ---------------------------------------------------------------------------
extern "C" void kernel_launch(void* const* d_in, const int* in_sizes, int n_in,
                              void* d_out, int out_size, void* d_ws, size_t ws_size,
                              hipStream_t stream)
{
    const float* latent = (const float*)d_in[0];   // [16384, 8, 32]
    const float* cb     = (const float*)d_in[1];   // [128, 32]

    float* policy = (float*)d_out;                        // 4,194,304 floats
    float* quant  = policy + (size_t)N_ROWS * EMB_DIM;    // 4,194,304 floats
    float* cbset  = quant  + (size_t)N_ROWS * EMB_DIM;    // 67,108,864 floats

    // 131072 rows / 16 rows-per-wave = 8192 waves; 8 waves (256 thr) per block
    vq_wmma_kernel<<<1024, 256, 0, stream>>>(latent, cb, policy, quant);

    // store-bandwidth-bound broadcast: 4096 blocks x 256 threads, grid-stride
    cb_broadcast_kernel<<<4096, 256, 0, stream>>>(cb, cbset);
}